// TritonKasminaLayer_22883585753475
// MI455X (gfx1250) — compile-verified
//
#include <hip/hip_runtime.h>
#include <cstdint>
#include <cstddef>

// Problem constants (from reference): BATCH=8192, HIDDEN=4096, NUM_SEEDS=32,
// CHUNK=128, NUM_BLUEPRINTS=10.
#define BATCH      8192
#define HIDDEN     4096
#define NUM_SEEDS  32
#define CHUNK      128

constexpr int TILE_ROWS = 16;                          // rows of x per TDM tile (tile = 16 x 128 f32 = 8KB)
constexpr int TPB_TILES = 4;                           // tiles per block -> 64 rows per block
constexpr int GRIDX     = BATCH / (TILE_ROWS * TPB_TILES); // 128 blocks along batch

typedef float        v4f __attribute__((ext_vector_type(4)));
typedef unsigned int v4u __attribute__((ext_vector_type(4)));
typedef int          v4i __attribute__((ext_vector_type(4)));
typedef int          v8i __attribute__((ext_vector_type(8)));

#if defined(__has_builtin)
#if __has_builtin(__builtin_amdgcn_tensor_load_to_lds) && __has_builtin(__builtin_amdgcn_s_wait_tensorcnt)
#define HAS_TDM 1
#endif
#endif
#ifndef HAS_TDM
#define HAS_TDM 0
#endif

// ---------------------------------------------------------------------------
// Main kernel: grid = (GRIDX, NUM_SEEDS). 256 threads = 8 wave32.
// blockIdx.y = seed (per-seed params uniform -> SALU); blockIdx.x = row tile group.
// x tiles are DMA'd by the Tensor Data Mover into double-buffered LDS; wave 0
// issues the DMA, TENSORcnt + workgroup barrier gate the consumers.
// ATOMIC=false: per-block partial sums written to workspace (deterministic).
// ATOMIC=true : accumulate directly into tsum/tsumsq with global float atomics.
// ---------------------------------------------------------------------------
template <bool ATOMIC>
__global__ __launch_bounds__(256) void kasmina_main(
    const float* __restrict__ x,
    const int*   __restrict__ lifecycle,
    const int*   __restrict__ bids,
    const int*   __restrict__ strats,
    const float* __restrict__ alphas,
    const float* __restrict__ bw,
    float* __restrict__ out,
    float* __restrict__ red0,   // ATOMIC ? tsum   : partial-sum array  [NUM_SEEDS*GRIDX]
    float* __restrict__ red1)   // ATOMIC ? tsumsq : partial-sq  array  [NUM_SEEDS*GRIDX]
{
  __shared__ __align__(16) float tile[2][TILE_ROWS * CHUNK];   // 16 KB ping-pong
  __shared__ float rs[8], rq[8];

  const int s    = blockIdx.y;
  const int tid  = threadIdx.x;
  const int wave = tid >> 5;
  const int lane = tid & 31;

  // Per-seed (block-uniform) parameters.
  const int   state  = lifecycle[s];
  const bool  active = (state >= 3) && (state <= 6);
  const int   strat  = strats[s];
  const float alpha  = alphas[s];
  const int   bid    = bids[s];

  // Per-lane blueprint weights for this seed's chunk (reused across all rows).
  const v4f wv = *reinterpret_cast<const v4f*>(bw + (size_t)bid * HIDDEN + s * CHUNK + lane * 4);

  const int rowBase = blockIdx.x * (TILE_ROWS * TPB_TILES);

#if HAS_TDM
  // Build a 2D tensor-DMA descriptor (D#) and issue TENSOR_LOAD_TO_LDS.
  // Group0: count=1, lds_addr, global_addr[56:0], type=2.
  // Group1: data_size=4B, tensor_dim0=HIDDEN, tensor_dim1=BATCH,
  //         tile_dim0=CHUNK, tile_dim1=TILE_ROWS, tensor_dim0_stride=HIDDEN.
  auto issue_tdm = [&](int t, int buf) {
    const uint64_t ga =
        (uint64_t)(uintptr_t)(x + (size_t)(rowBase + t * TILE_ROWS) * HIDDEN + s * CHUNK);
    const uint32_t laddr = (uint32_t)(uintptr_t)&tile[buf][0];  // LDS byte offset
    v4u g0;
    g0.x = 1u;                                                   // count=1, user mode
    g0.y = laddr;                                                // lds_addr
    g0.z = (uint32_t)(ga & 0xFFFFFFFFu);                         // global_addr[31:0]
    g0.w = (uint32_t)((ga >> 32) & 0x01FFFFFFu) | (2u << 30);    // global_addr[56:32] | type=2
    v8i g1;
    g1[0] = (2 << 16);                                           // data_size = 4 bytes
    g1[1] = (int)(((unsigned)HIDDEN & 0xFFFFu) << 16);           // tensor_dim0[15:0]
    g1[2] = (int)(((unsigned)HIDDEN >> 16) |
                  (((unsigned)BATCH & 0xFFFFu) << 16));          // tensor_dim0[31:16] | tensor_dim1[15:0]
    g1[3] = (int)(((unsigned)BATCH >> 16) |
                  ((unsigned)CHUNK << 16));                      // tensor_dim1[31:16] | tile_dim0
    g1[4] = TILE_ROWS;                                           // tile_dim1 (tile_dim2 = 0)
    g1[5] = HIDDEN;                                              // tensor_dim0_stride[31:0]
    g1[6] = 0;                                                   // stride0[47:32] | stride1[15:0]
    g1[7] = 0;
    v4i gz = {};
#if __clang_major__ >= 23
    v8i gx = {};
    __builtin_amdgcn_tensor_load_to_lds(g0, g1, gz, gz, gx, 0);
#else
    __builtin_amdgcn_tensor_load_to_lds(g0, g1, gz, gz, 0);
#endif
  };
  if (wave == 0) issue_tdm(0, 0);   // prologue: start tile 0 DMA
#endif

  float acc = 0.0f, accsq = 0.0f;

  for (int t = 0; t < TPB_TILES; ++t) {
    const int buf = t & 1;
#if HAS_TDM
    if (wave == 0) {
      if (t + 1 < TPB_TILES) {
        issue_tdm(t + 1, (t + 1) & 1);             // overlap next DMA with compute
        __builtin_amdgcn_s_wait_tensorcnt(1);      // tile t complete (in-order TDM)
      } else {
        __builtin_amdgcn_s_wait_tensorcnt(0);
      }
    }
    __syncthreads();                               // publish LDS tile to all waves
#else
    // Fallback: cooperative global->LDS staging.
    for (int r = wave; r < TILE_ROWS; r += 8) {
      v4f v = *reinterpret_cast<const v4f*>(
          x + (size_t)(rowBase + t * TILE_ROWS + r) * HIDDEN + s * CHUNK + lane * 4);
      *reinterpret_cast<v4f*>(&tile[buf][r * CHUNK + lane * 4]) = v;
    }
    __syncthreads();
#endif

    for (int r = wave; r < TILE_ROWS; r += 8) {
      const v4f xv = *reinterpret_cast<const v4f*>(&tile[buf][r * CHUNK + lane * 4]);
      v4f o;
      if (!active) {
        o = xv;
      } else if (strat == 0) {
        o = (alpha * wv) * xv + (1.0f - alpha) * xv;
      } else if (strat == 1) {
        o = wv * xv;
      } else {
        o = xv + wv;
      }
      acc   += xv.x + xv.y + xv.z + xv.w;
      accsq += xv.x * xv.x + xv.y * xv.y + xv.z * xv.z + xv.w * xv.w;

      const size_t row = (size_t)(rowBase + t * TILE_ROWS + r);
      // Streaming output, no reuse -> non-temporal 128-bit store.
      __builtin_nontemporal_store(
          o, reinterpret_cast<v4f*>(out + row * HIDDEN + s * CHUNK + lane * 4));
    }
    __syncthreads();   // all waves done reading buf before it is DMA-overwritten
  }

  // Block reduction of the per-thread x-statistics (seed is block-uniform).
  for (int off = 16; off > 0; off >>= 1) {
    acc   += __shfl_down(acc, off, 32);
    accsq += __shfl_down(accsq, off, 32);
  }
  if (lane == 0) { rs[wave] = acc; rq[wave] = accsq; }
  __syncthreads();
  if (tid == 0) {
    float a = 0.0f, b = 0.0f;
#pragma unroll
    for (int i = 0; i < 8; ++i) { a += rs[i]; b += rq[i]; }
    if (ATOMIC) {
      atomicAdd(&red0[s], a);
      atomicAdd(&red1[s], b);
    } else {
      red0[s * GRIDX + blockIdx.x] = a;
      red1[s * GRIDX + blockIdx.x] = b;
    }
  }
}

// ---------------------------------------------------------------------------
// Stage 2 (deterministic path): reduce GRIDX partials per seed.
// ---------------------------------------------------------------------------
__global__ __launch_bounds__(128) void kasmina_reduce(
    const float* __restrict__ ws, float* __restrict__ tsum,
    float* __restrict__ tsumsq, float* __restrict__ tcount)
{
  const int s = blockIdx.x, tid = threadIdx.x;
  float a = 0.0f, b = 0.0f;
  for (int i = tid; i < GRIDX; i += blockDim.x) {
    a += ws[s * GRIDX + i];
    b += ws[NUM_SEEDS * GRIDX + s * GRIDX + i];
  }
  for (int off = 16; off > 0; off >>= 1) {
    a += __shfl_down(a, off, 32);
    b += __shfl_down(b, off, 32);
  }
  __shared__ float sa[4], sb[4];
  const int w = tid >> 5, l = tid & 31;
  if (l == 0) { sa[w] = a; sb[w] = b; }
  __syncthreads();
  if (tid == 0) {
    tsum[s]   = sa[0] + sa[1] + sa[2] + sa[3];
    tsumsq[s] = sb[0] + sb[1] + sb[2] + sb[3];
    tcount[s] = (float)((size_t)BATCH * CHUNK);   // 1048576.0f
  }
}

// Atomic fallback: zero tsum/tsumsq, set tcount.
__global__ void kasmina_init_stats(float* __restrict__ stats)
{
  const int i = threadIdx.x;
  if (i < 2 * NUM_SEEDS)      stats[i] = 0.0f;
  else if (i < 3 * NUM_SEEDS) stats[i] = (float)((size_t)BATCH * CHUNK);
}

// ---------------------------------------------------------------------------
extern "C" void kernel_launch(void* const* d_in, const int* in_sizes, int n_in,
                              void* d_out, int out_size, void* d_ws, size_t ws_size,
                              hipStream_t stream)
{
  (void)in_sizes; (void)n_in; (void)out_size;

  const float* x      = (const float*)d_in[0];
  const int*   ls     = (const int*)d_in[1];
  const int*   bid    = (const int*)d_in[2];
  const int*   strat  = (const int*)d_in[3];
  const float* alpha  = (const float*)d_in[4];
  const float* bw     = (const float*)d_in[5];

  float* out    = (float*)d_out;
  float* tsum   = out + (size_t)BATCH * HIDDEN;
  float* tsumsq = tsum + NUM_SEEDS;
  float* tcount = tsumsq + NUM_SEEDS;

  dim3 grid(GRIDX, NUM_SEEDS, 1);
  const size_t need = (size_t)2 * NUM_SEEDS * GRIDX * sizeof(float);  // 32 KB

  if (d_ws != nullptr && ws_size >= need) {
    float* ws = (float*)d_ws;
    kasmina_main<false><<<grid, 256, 0, stream>>>(x, ls, bid, strat, alpha, bw, out,
                                                  ws, ws + (size_t)NUM_SEEDS * GRIDX);
    kasmina_reduce<<<NUM_SEEDS, 128, 0, stream>>>(ws, tsum, tsumsq, tcount);
  } else {
    kasmina_init_stats<<<1, 3 * NUM_SEEDS, 0, stream>>>(tsum);
    kasmina_main<true><<<grid, 256, 0, stream>>>(x, ls, bid, strat, alpha, bw, out,
                                                 tsum, tsumsq);
  }
}